// SingleStreamMemoryBank_90941637525850
// MI455X (gfx1250) — compile-verified
//
#include <hip/hip_runtime.h>
#include <cstddef>
#include <cstdint>

// SingleStreamMemoryBank fused kernel for gfx1250 (MI455X).
// B=4096 rows, K=64 slots, D=1024. One workgroup per row; bank row cached in
// LDS (async-to-LDS), dots + both-branch weighted sums via V_WMMA_F32_16X16X4_F32.

#define NB       4096
#define NK       64
#define ND       1024
#define SSTRIDE  1028            // padded LDS row stride (floats): 16B aligned, conflict-free
#define NTHREADS 512             // 16 waves of 32
#define THRESH   0.8f
#define EPSN     1e-12f

typedef float v2f __attribute__((ext_vector_type(2)));
typedef float v8f __attribute__((ext_vector_type(8)));
typedef int   v4i __attribute__((ext_vector_type(4)));

#if defined(__has_builtin)
#if __has_builtin(__builtin_amdgcn_global_load_async_to_lds_b128)
#define HAVE_ASYNC_LDS 1
#endif
#endif

// LDS float layout (carved from dynamic shared):
//  S: NK*SSTRIDE | qs: ND | its: ND | pq/pi/pn: 3*256 | qdot/idot/nrm2: 3*64
//  aE/aF: 2*64 | wr: 48 | sc: 8
#define SMEM_FLOATS (NK*SSTRIDE + 2*ND + 3*256 + 3*64 + 2*64 + 48 + 8)

__global__ __launch_bounds__(NTHREADS) void ssmb_row_kernel(
    const float* __restrict__ query, const float* __restrict__ item,
    const float* __restrict__ bank, float* __restrict__ out_e,
    float* __restrict__ out_f, float* __restrict__ bestsim)
{
    extern __shared__ float smem[];
    const int b    = blockIdx.x;
    const int tid  = threadIdx.x;
    const int lane = tid & 31;
    const int wave = tid >> 5;
    const int m    = lane & 15;
    const bool hi  = lane >= 16;
    const float selm = (m < 2) ? 1.f : 0.f;   // zero A rows M>=2 without predication

    float* S    = smem;
    float* qs   = S + NK * SSTRIDE;
    float* its  = qs + ND;
    float* pq   = its + ND;
    float* pi   = pq + 256;
    float* pn   = pi + 256;
    float* qdot = pn + 256;
    float* idot = qdot + 64;
    float* nrm2 = idot + 64;
    float* aE   = nrm2 + 64;
    float* aF   = aE + 64;
    float* wr   = aF + 64;
    float* sc   = wr + 48;

    const float* bankRow = bank + (size_t)b * (NK * ND);

    // ---- Stage A: stage bank row into LDS (padded stride), plus query/item ----
#if HAVE_ASYNC_LDS
    for (int g = tid; g < NK * (ND / 4); g += NTHREADS) {
        const int slot = g >> 8;              // 256 quads per slot
        const int d4   = (g & 255) << 2;
        const float* gp = bankRow + slot * ND + d4;
        float*       lp = S + slot * SSTRIDE + d4;
        __builtin_amdgcn_global_load_async_to_lds_b128(
            (__attribute__((address_space(1))) v4i*)gp,
            (__attribute__((address_space(3))) v4i*)lp, 0, 0);
    }
#else
    for (int g = tid; g < NK * (ND / 4); g += NTHREADS) {
        const int slot = g >> 8;
        const int d4   = (g & 255) << 2;
        const float4 v = *(const float4*)(bankRow + slot * ND + d4);
        float* lp = S + slot * SSTRIDE + d4;
        lp[0] = v.x; lp[1] = v.y; lp[2] = v.z; lp[3] = v.w;
    }
#endif
    for (int d = tid; d < ND; d += NTHREADS) {
        qs[d]  = query[(size_t)b * ND + d];
        its[d] = item[(size_t)b * ND + d];
    }
#if HAVE_ASYNC_LDS
#if __has_builtin(__builtin_amdgcn_s_wait_asynccnt)
    __builtin_amdgcn_s_wait_asynccnt(0);
#else
    asm volatile("s_wait_asynccnt 0" ::: "memory");
#endif
#endif
    __syncthreads();

    // ---- Stage B1: scalar stats ||q||^2, ||item||^2, q.item (deterministic) ----
    {
        float p0 = 0.f, p1 = 0.f, p2 = 0.f;
        for (int d = tid; d < ND; d += NTHREADS) {
            const float qv = qs[d], iv = its[d];
            p0 += qv * qv; p1 += iv * iv; p2 += qv * iv;
        }
        for (int off = 16; off > 0; off >>= 1) {
            p0 += __shfl_down(p0, off, 32);
            p1 += __shfl_down(p1, off, 32);
            p2 += __shfl_down(p2, off, 32);
        }
        if (lane == 0) { wr[wave] = p0; wr[16 + wave] = p1; wr[32 + wave] = p2; }
    }

    // ---- Stage B2: dots via WMMA. wave -> (slot tile of 16, d-segment of 256) ----
    // C[M][N] += A[M][kk] * B[kk][N];  A: M=0 -> query, M=1 -> item over 4 d's;
    // B[kk][N] = S[slot0+N][d0+kk].  64 chunks of K=4, 4 independent accumulators.
    {
        const int tile  = wave & 3;
        const int dseg  = wave >> 2;
        const int slot0 = tile << 4;
        const float* Srow = S + (slot0 + m) * SSTRIDE;
        const float* ap   = (m == 1) ? its : qs;   // lanes m>=2 broadcast qs, zeroed via selm
        const int dbase   = (dseg << 8) + (hi ? 2 : 0);
        v8f acc[4] = {{}, {}, {}, {}};
        float np = 0.f;
        for (int c = 0; c < 16; ++c) {
#pragma unroll
            for (int j = 0; j < 4; ++j) {
                const int dd = dbase + (c << 2) + (j << 6);
                const float a0 = ap[dd], a1 = ap[dd + 1];
                v2f a;  a.x = a0 * selm;  a.y = a1 * selm;
                v2f bm; bm.x = Srow[dd];  bm.y = Srow[dd + 1];
                np += bm.x * bm.x + bm.y * bm.y;
                acc[j] = __builtin_amdgcn_wmma_f32_16x16x4_f32(
                    false, a, false, bm, (short)0, acc[j], false, false);
            }
        }
        const v8f accT = (acc[0] + acc[1]) + (acc[2] + acc[3]);
        np += __shfl_xor(np, 16, 32);    // combine low/high K-halves of the norm
        if (lane < 16) {
            const int s = slot0 + lane;  // C: VGPR j, lanes 0-15 => M=j, N=lane
            pq[dseg * 64 + s] = accT[0]; // M=0 row: query dot partial
            pi[dseg * 64 + s] = accT[1]; // M=1 row: item dot partial
            pn[dseg * 64 + s] = np;
        }
    }
    __syncthreads();

    // ---- Stage C1: ordered combine of d-segment partials ----
    if (tid < NK) {
        float q = 0.f, i = 0.f, n = 0.f;
        for (int s = 0; s < 4; ++s) {
            q += pq[s * 64 + tid]; i += pi[s * 64 + tid]; n += pn[s * 64 + tid];
        }
        qdot[tid] = q; idot[tid] = i; nrm2[tid] = n;
    }
    __syncthreads();

    // ---- Stage C2: per-row logic on one thread (argmax, both softmaxes, alphas) ----
    if (tid == 0) {
        float q2 = 0.f, i2 = 0.f, qi = 0.f;
        for (int w = 0; w < 16; ++w) { q2 += wr[w]; i2 += wr[16 + w]; qi += wr[32 + w]; }
        const float nq = fmaxf(sqrtf(q2), EPSN);
        const float ni = fmaxf(sqrtf(i2), EPSN);

        int bidx = 0; float bs = -3.4e38f;
        for (int k = 0; k < NK; ++k) {
            const float ns = fmaxf(sqrtf(nrm2[k]), EPSN);
            const float s  = idot[k] / (ni * ns);
            if (s > bs) { bs = s; bidx = k; }   // strict '>' keeps first max (argmax)
        }
        bestsim[b] = bs;

        // EMA branch: slot bidx replaced by 0.5*(slot+item)
        for (int k = 0; k < NK; ++k) {
            const float ns = fmaxf(sqrtf(nrm2[k]), EPSN);
            aE[k] = qdot[k] / (nq * ns);
        }
        {
            const float qe  = 0.5f * (qdot[bidx] + qi);
            const float n2e = 0.25f * (nrm2[bidx] + 2.f * idot[bidx] + i2);
            aE[bidx] = qe / (nq * fmaxf(sqrtf(n2e), EPSN));
        }
        // FIFO branch: new_bank[k] = bank[k+1] (k<63), new_bank[63] = item
        for (int k = 0; k < NK - 1; ++k) {
            const float ns = fmaxf(sqrtf(nrm2[k + 1]), EPSN);
            aF[k] = qdot[k + 1] / (nq * ns);
        }
        aF[NK - 1] = qi / (nq * ni);

        // softmax both (in place)
        for (int v = 0; v < 2; ++v) {
            float* a = v ? aF : aE;
            float mx = a[0];
            for (int k = 1; k < NK; ++k) mx = fmaxf(mx, a[k]);
            float sum = 0.f;
            for (int k = 0; k < NK; ++k) { a[k] = expf(a[k] - mx); sum += a[k]; }
            const float inv = 1.f / sum;
            for (int k = 0; k < NK; ++k) a[k] *= inv;
        }
        // coefficients on ORIGINAL bank slots + item
        const float betaE = 0.5f * aE[bidx];
        aE[bidx] = betaE;
        const float betaF = aF[NK - 1];
        for (int j = NK - 1; j >= 1; --j) aF[j] = aF[j - 1];
        aF[0] = 0.f;
        sc[0] = betaE; sc[1] = betaF;
    }
    __syncthreads();

    // ---- Stage D: both candidate outputs via WMMA ----
    // C[M][N] += A[M][kk]*B[kk][N];  A: M=0 -> aE, M=1 -> aF over 4 slots;
    // B[kk][N] = S[k0+kk][d0+N].  Each wave: 4 d-chunks of 16, shared A operand.
    {
        const float betaE = sc[0], betaF = sc[1];
        const float* ap = (m == 1) ? aF : aE;     // lanes m>=2 broadcast aE, zeroed via selm
        const int dbase = wave << 6;              // 64 d's per wave
        v8f acc[4] = {{}, {}, {}, {}};
        for (int k0 = 0; k0 < NK; k0 += 4) {
            const int kb = k0 + (hi ? 2 : 0);
            const float a0 = ap[kb], a1 = ap[kb + 1];
            v2f a; a.x = a0 * selm; a.y = a1 * selm;
            const float* Sb0 = S + kb * SSTRIDE + dbase + m;
#pragma unroll
            for (int j = 0; j < 4; ++j) {
                v2f bm; bm.x = Sb0[j * 16]; bm.y = Sb0[SSTRIDE + j * 16];
                acc[j] = __builtin_amdgcn_wmma_f32_16x16x4_f32(
                    false, a, false, bm, (short)0, acc[j], false, false);
            }
        }
        if (lane < 16) {
#pragma unroll
            for (int j = 0; j < 4; ++j) {
                const int d = dbase + j * 16 + lane;
                const float itv = its[d];
                out_e[(size_t)b * ND + d] = acc[j][0] + betaE * itv;  // M=0 row
                out_f[(size_t)b * ND + d] = acc[j][1] + betaF * itv;  // M=1 row
            }
        }
    }
}

__global__ void ssmb_gate_kernel(const float* __restrict__ bestsim, int* __restrict__ flag)
{
    __shared__ float red[1024];
    const int t = threadIdx.x;
    red[t] = bestsim[t] + bestsim[t + 1024] + bestsim[t + 2048] + bestsim[t + 3072];
    __syncthreads();
    for (int off = 512; off > 0; off >>= 1) {
        if (t < off) red[t] += red[t + off];
        __syncthreads();
    }
    if (t == 0) flag[0] = (red[0] * (1.0f / (float)NB) >= THRESH) ? 1 : 0;
}

__global__ void ssmb_select_kernel(const float* __restrict__ oe, const float* __restrict__ of,
                                   const int* __restrict__ flag, float* __restrict__ out, int n)
{
    const float* src = flag[0] ? oe : of;
    for (int i = blockIdx.x * blockDim.x + threadIdx.x; i < n; i += gridDim.x * blockDim.x)
        out[i] = src[i];
}

extern "C" void kernel_launch(void* const* d_in, const int* in_sizes, int n_in,
                              void* d_out, int out_size, void* d_ws, size_t ws_size,
                              hipStream_t stream)
{
    (void)in_sizes; (void)n_in; (void)ws_size;
    const float* query = (const float*)d_in[0];
    const float* item  = (const float*)d_in[1];
    const float* bank  = (const float*)d_in[2];
    float* out = (float*)d_out;

    float* out_e   = (float*)d_ws;
    float* out_f   = out_e + (size_t)NB * ND;
    float* bestsim = out_f + (size_t)NB * ND;
    int*   flag    = (int*)(bestsim + NB);

    const size_t smem_bytes = (size_t)SMEM_FLOATS * sizeof(float);  // ~276 KB (<=320 KB/WGP)
    (void)hipFuncSetAttribute((const void*)ssmb_row_kernel,
                              hipFuncAttributeMaxDynamicSharedMemorySize, (int)smem_bytes);

    ssmb_row_kernel<<<NB, NTHREADS, smem_bytes, stream>>>(query, item, bank,
                                                          out_e, out_f, bestsim);
    ssmb_gate_kernel<<<1, 1024, 0, stream>>>(bestsim, flag);
    ssmb_select_kernel<<<4096, 256, 0, stream>>>(out_e, out_f, flag, out, out_size);
}